// EGATBlock_63977832841244
// MI455X (gfx1250) — compile-verified
//
#include <hip/hip_runtime.h>
#include <hip/hip_bf16.h>

typedef float v2f __attribute__((ext_vector_type(2)));
typedef float v8f __attribute__((ext_vector_type(8)));

#define IN_NF 128   // input node feature dim
#define IN_EF 32    // input edge feature dim
#define HOE   64    // H * OUT_E
#define HON   128   // H * OUT_N
#define NHEAD 4

// ---- ordered-uint encoding for float atomic max ----
__device__ __forceinline__ unsigned enc_f32(float f) {
    unsigned b = __float_as_uint(f);
    return (b & 0x80000000u) ? ~b : (b | 0x80000000u);
}
__device__ __forceinline__ float dec_f32(unsigned u) {
    unsigned b = (u & 0x80000000u) ? (u & 0x7FFFFFFFu) : ~u;
    return __uint_as_float(b);
}
#define ENC_NEG_INF 0x007FFFFFu   // enc_f32(-INFINITY)

// ---------------------------------------------------------------------------
// K0: zero h_out, init segment-max (encoded -inf) and segment-sum buffers
// ---------------------------------------------------------------------------
__global__ __launch_bounds__(256) void egat_init_kernel(
    float* __restrict__ h_out, unsigned* __restrict__ m_enc,
    float* __restrict__ s, int N)
{
    int idx = blockIdx.x * 256 + threadIdx.x;
    if (idx < N * HON) h_out[idx] = 0.0f;
    if (idx < N * NHEAD) { m_enc[idx] = ENC_NEG_INF; s[idx] = 0.0f; }
}

// ---------------------------------------------------------------------------
// K1: node projections  f_ni = X@W_ni, f_nj = X@W_nj, h = X@W_node
// 16-node row tile per block; A staged in LDS; 8 waves x grid.y=2 cover the
// 16 column tiles of the concatenated [64 | 64 | 128] output.
// Per-wave GEMM templated on the (compile-time) output stride so W loads and
// C stores use immediate offsets; full-tile fast path avoids per-row guards.
// ---------------------------------------------------------------------------
#define LDA (IN_NF + 2)   // padded LDS stride (conflict avoidance, 8B aligned)

template <int LDW>
__device__ __forceinline__ void wmma_col_gemm_store(
    const float* __restrict__ Atile,   // LDS, 16 x IN_NF (stride LDA)
    const float* __restrict__ W,       // [IN_NF, LDW] row-major
    float*       __restrict__ out,     // [*, LDW]
    int node_base, int col, int N, int hl, int l16)
{
    v8f acc = {};
    const float* wp = W + (size_t)(2 * hl) * LDW + col + l16;
    const float* ap = Atile + l16 * LDA + 2 * hl;
#pragma unroll 4
    for (int k0 = 0; k0 < IN_NF; k0 += 4) {
        v2f a, b;
        a.x = ap[0];
        a.y = ap[1];
        b.x = wp[0];
        b.y = wp[LDW];
        ap += 4;
        wp += 4 * LDW;
        acc = __builtin_amdgcn_wmma_f32_16x16x4_f32(
            false, a, false, b, (short)0, acc, false, false);
    }

    float* op = out + (size_t)(node_base + 8 * hl) * LDW + col + l16;
    if (node_base + 15 < N) {          // uniform fast path: no guards
#pragma unroll
        for (int r = 0; r < 8; ++r) { op[0] = acc[r]; op += LDW; }
    } else {
#pragma unroll
        for (int r = 0; r < 8; ++r) {
            if (node_base + 8 * hl + r < N) op[0] = acc[r];
            op += LDW;
        }
    }
}

__global__ __launch_bounds__(256) void egat_node_proj_kernel(
    const float* __restrict__ n_feats,
    const float* __restrict__ W_ni,
    const float* __restrict__ W_nj,
    const float* __restrict__ W_node,
    float* __restrict__ f_ni,
    float* __restrict__ f_nj,
    float* __restrict__ h,
    int N)
{
    __shared__ float Atile[16 * LDA];

    const int node_base = blockIdx.x * 16;
    // cooperative, coalesced stage of the 16x128 A tile
    for (int i = threadIdx.x; i < 16 * IN_NF; i += 256) {
        int r = i >> 7;          // /IN_NF
        int k = i & (IN_NF - 1);
        int node = node_base + r;
        Atile[r * LDA + k] = (node < N) ? n_feats[(size_t)node * IN_NF + k] : 0.0f;
    }
    __syncthreads();

    const int wave = threadIdx.x >> 5;
    const int lane = threadIdx.x & 31;
    const int hl   = lane >> 4;      // 0: K=0,1 rows   1: K=2,3 rows
    const int l16  = lane & 15;
    const int ct   = blockIdx.y * 8 + wave;   // 0..15 column tile (wave-uniform)

    if (ct < 4) {
        wmma_col_gemm_store<HOE>(Atile, W_ni,   f_ni, node_base, ct * 16,       N, hl, l16);
    } else if (ct < 8) {
        wmma_col_gemm_store<HOE>(Atile, W_nj,   f_nj, node_base, (ct - 4) * 16, N, hl, l16);
    } else {
        wmma_col_gemm_store<HON>(Atile, W_node, h,    node_base, (ct - 8) * 16, N, hl, l16);
    }
}

// ---------------------------------------------------------------------------
// K2: edge GEMM f_fij (WMMA) fused with gather + bias + leaky_relu -> f_out,
// attention logit (16-lane shuffle dot) -> e_ws, segment max -> m_enc.
// Block = 4 waves (one per head) over a 16-edge tile.
// ---------------------------------------------------------------------------
#define LDE (IN_EF + 2)
__global__ __launch_bounds__(128) void egat_edge_kernel(
    const float* __restrict__ e_feats,
    const int*   __restrict__ src,
    const int*   __restrict__ dst,
    const float* __restrict__ W_fij,
    const float* __restrict__ bias_e,
    const float* __restrict__ attn,
    const float* __restrict__ f_ni,
    const float* __restrict__ f_nj,
    float*    __restrict__ f_out,
    float*    __restrict__ e_ws,
    unsigned* __restrict__ m_enc,
    int E)
{
    __shared__ float Etile[16 * LDE];

    const int ebase = blockIdx.x * 16;
    for (int i = threadIdx.x; i < 16 * IN_EF; i += 128) {
        int r = i >> 5;           // /IN_EF
        int k = i & (IN_EF - 1);
        int e = ebase + r;
        Etile[r * LDE + k] = (e < E) ? e_feats[(size_t)e * IN_EF + k] : 0.0f;
    }
    __syncthreads();

    const int head = threadIdx.x >> 5;   // wave id == head
    const int lane = threadIdx.x & 31;
    const int hl   = lane >> 4;
    const int l16  = lane & 15;
    const int col  = head * 16;          // 16 cols == one head

    v8f acc = {};
    {
        const float* wp = W_fij + (size_t)(2 * hl) * HOE + col + l16;
        const float* ap = Etile + l16 * LDE + 2 * hl;
#pragma unroll
        for (int k0 = 0; k0 < IN_EF; k0 += 4) {
            v2f a, b;
            a.x = ap[0];
            a.y = ap[1];
            b.x = wp[0];
            b.y = wp[HOE];
            ap += 4;
            wp += 4 * HOE;
            acc = __builtin_amdgcn_wmma_f32_16x16x4_f32(
                false, a, false, b, (short)0, acc, false, false);
        }
    }

    const float attw = attn[head * 16 + l16];
    const float bia  = bias_e[col + l16];
    const bool  full = (ebase + 15 < E);   // uniform

    float* fp = f_out + (size_t)(ebase + 8 * hl) * HOE + col + l16;

#pragma unroll
    for (int r = 0; r < 8; ++r) {
        int e = ebase + r + 8 * hl;      // lanes 0-15: row r; lanes 16-31: row r+8
        if (full || e < E) {
            int sv = src[e], dv = dst[e];
            float v = acc[r]
                    + f_ni[(size_t)sv * HOE + col + l16]
                    + f_nj[(size_t)dv * HOE + col + l16]
                    + bia;
            v = (v > 0.0f) ? v : 0.01f * v;               // leaky_relu
            fp[0] = v;

            float p = v * attw;                            // attention logit dot
            p += __shfl_xor(p, 1, 32);                     // masks <16 keep the
            p += __shfl_xor(p, 2, 32);                     // two 16-lane halves
            p += __shfl_xor(p, 4, 32);                     // (distinct edges)
            p += __shfl_xor(p, 8, 32);                     // separate
            if (l16 == 0) {
                e_ws[(size_t)e * NHEAD + head] = p;
                atomicMax(&m_enc[(size_t)dv * NHEAD + head], enc_f32(p));
            }
        }
        fp += HOE;
    }
}

// ---------------------------------------------------------------------------
// K3: ex = exp(e - m[dst]);  s[dst] += ex  (per edge per head)
// ---------------------------------------------------------------------------
__global__ __launch_bounds__(256) void egat_softmax_kernel(
    const int* __restrict__ dst,
    const unsigned* __restrict__ m_enc,
    float* __restrict__ e_ws,
    float* __restrict__ s,
    int E)
{
    int idx = blockIdx.x * 256 + threadIdx.x;
    if (idx >= E * NHEAD) return;
    int edge = idx >> 2;
    int hh   = idx & 3;
    int d    = dst[edge];
    float m  = dec_f32(m_enc[(size_t)d * NHEAD + hh]);
    float ex = __expf(e_ws[idx] - m);
    e_ws[idx] = ex;
    unsafeAtomicAdd(&s[(size_t)d * NHEAD + hh], ex);
}

// ---------------------------------------------------------------------------
// K4: a = ex / s[dst];  h_out[dst, :] += h[src, :] * a   (thread per edge,col)
// 128 consecutive threads cover one edge's full 512B row -> coalesced.
// ---------------------------------------------------------------------------
__global__ __launch_bounds__(256) void egat_scatter_kernel(
    const int* __restrict__ src,
    const int* __restrict__ dst,
    const float* __restrict__ h,
    const float* __restrict__ ex_ws,
    const float* __restrict__ s,
    float* __restrict__ h_out,
    int E)
{
    int tid = blockIdx.x * 256 + threadIdx.x;
    if (tid >= E * HON) return;
    int edge = tid >> 7;        // /HON
    int c    = tid & (HON - 1);
    int hh   = c >> 5;          // /OUT_N
    int dv = dst[edge], sv = src[edge];
    float a = ex_ws[(size_t)edge * NHEAD + hh] / s[(size_t)dv * NHEAD + hh];
    unsafeAtomicAdd(&h_out[(size_t)dv * HON + c], h[(size_t)sv * HON + c] * a);
}

// ---------------------------------------------------------------------------
extern "C" void kernel_launch(void* const* d_in, const int* in_sizes, int n_in,
                              void* d_out, int out_size, void* d_ws, size_t ws_size,
                              hipStream_t stream) {
    const float* n_feats = (const float*)d_in[0];
    const float* e_feats = (const float*)d_in[1];
    const int*   src     = (const int*)  d_in[2];
    const int*   dst     = (const int*)  d_in[3];
    const float* W_ni    = (const float*)d_in[4];
    const float* W_nj    = (const float*)d_in[5];
    const float* W_fij   = (const float*)d_in[6];
    const float* bias_e  = (const float*)d_in[7];
    const float* attn    = (const float*)d_in[8];
    const float* W_node  = (const float*)d_in[9];

    const int N = in_sizes[0] / IN_NF;
    const int E = in_sizes[2];

    // outputs: (h_out [N,H,OUT_N], f_out [E,H,OUT_E]) concatenated
    float* h_out = (float*)d_out;
    float* f_out = h_out + (size_t)N * HON;

    // workspace layout
    float* f_ni = (float*)d_ws;
    float* f_nj = f_ni + (size_t)N * HOE;
    float* h    = f_nj + (size_t)N * HOE;
    float* e_ws = h    + (size_t)N * HON;
    unsigned* m_enc = (unsigned*)(e_ws + (size_t)E * NHEAD);
    float* s    = (float*)(m_enc + (size_t)N * NHEAD);

    const int nt16 = (N + 15) / 16;
    const int et16 = (E + 15) / 16;

    egat_init_kernel<<<(N * HON + 255) / 256, 256, 0, stream>>>(h_out, m_enc, s, N);

    egat_node_proj_kernel<<<dim3(nt16, 2), 256, 0, stream>>>(
        n_feats, W_ni, W_nj, W_node, f_ni, f_nj, h, N);

    egat_edge_kernel<<<et16, 128, 0, stream>>>(
        e_feats, src, dst, W_fij, bias_e, attn, f_ni, f_nj,
        f_out, e_ws, m_enc, E);

    egat_softmax_kernel<<<(E * NHEAD + 255) / 256, 256, 0, stream>>>(
        dst, m_enc, e_ws, s, E);

    egat_scatter_kernel<<<(E * HON + 255) / 256, 256, 0, stream>>>(
        src, dst, h, e_ws, s, h_out, E);
}